// KimiLinearKDADecode_42803644072882
// MI455X (gfx1250) — compile-verified
//
#include <hip/hip_runtime.h>
#include <math.h>

typedef __attribute__((ext_vector_type(2))) float v2f;
typedef __attribute__((ext_vector_type(8))) float v8f;

#define HK  32
#define HV  32
#define DD  128
#define BB  128
#define QKV ((2*HK + HV) * DD)   // 12288

__device__ __forceinline__ float fast_sigmoid(float x) {
    return 1.0f / (1.0f + __expf(-x));
}
__device__ __forceinline__ float fast_silu(float x) {
    return x * fast_sigmoid(x);
}
__device__ __forceinline__ float fast_softplus(float x) {
    // matches jax.nn.softplus = log1p(exp(x)) with overflow guard
    return (x > 20.0f) ? x : log1pf(__expf(x));
}

__global__ __launch_bounds__(128)
void kda_decode_kernel(const float* __restrict__ mixed_qkv,
                       const float* __restrict__ forget_gate,
                       const float* __restrict__ beta,
                       const float* __restrict__ conv_state,
                       const float* __restrict__ conv_weights,
                       const float* __restrict__ ssm_state,
                       const float* __restrict__ A_log,
                       const float* __restrict__ dt_bias,
                       float* __restrict__ out)
{
    const int bh   = blockIdx.x;          // 0..4095
    const int b    = bh / HV;
    const int h    = bh % HV;
    const int tid  = threadIdx.x;         // 0..127
    const int lane = tid & 31;            // wave32 lane
    const int wave = tid >> 5;            // 0..3

    __shared__ float s_qe[DD];            // qhat * exp(g)
    __shared__ float s_ke[DD];            // khat * exp(g)
    __shared__ float s_v[DD];
    __shared__ float s_red[4][4];
    __shared__ float s_qS[4][DD];         // per-wave partial q·Sg
    __shared__ float s_kS[4][DD];         // per-wave partial k·Sg

    // ---------------- Phase 1: per-dim preprocessing (tid == d) -------------
    const int d = tid;

    // conv (width 4: 3 state taps + current input) + SiLU, per channel
    auto conv_silu = [&](int c) -> float {
        const float* cs = conv_state + ((size_t)b * QKV + c) * 3;
        const float* w  = conv_weights + (size_t)c * 4;
        float x = cs[0]*w[0] + cs[1]*w[1] + cs[2]*w[2]
                + mixed_qkv[(size_t)b * QKV + c] * w[3];
        return fast_silu(x);
    };
    const float qraw = conv_silu( h          * DD + d);
    const float kraw = conv_silu((HK  + h)   * DD + d);
    const float vval = conv_silu((2*HK + h)  * DD + d);

    const float fg  = forget_gate[(size_t)b * (HV*DD) + h*DD + d];
    const float dtb = dt_bias[h*DD + d];
    const float g   = -__expf(A_log[h]) * fast_softplus(fg + dtb);
    const float eg  = __expf(g);

    // wave32 tree reductions: sum q^2, sum k^2, dot(q,k)
    float r0 = qraw*qraw, r1 = kraw*kraw, r2 = qraw*kraw;
    #pragma unroll
    for (int m = 16; m >= 1; m >>= 1) {
        r0 += __shfl_xor(r0, m, 32);
        r1 += __shfl_xor(r1, m, 32);
        r2 += __shfl_xor(r2, m, 32);
    }
    if (lane == 0) { s_red[wave][0] = r0; s_red[wave][1] = r1; s_red[wave][2] = r2; }
    __syncthreads();
    float sq = 0.f, sk = 0.f, sqk = 0.f;
    #pragma unroll
    for (int w2 = 0; w2 < 4; ++w2) {
        sq  += s_red[w2][0];
        sk  += s_red[w2][1];
        sqk += s_red[w2][2];
    }
    const float rsq    = rsqrtf(sq + 1e-6f) * 0.08838834764831845f; // includes D^-0.5
    const float rsk    = rsqrtf(sk + 1e-6f);
    const float qk_dot = sqk * rsq * rsk;                           // dot(qhat, khat)

    s_qe[d] = (qraw * rsq) * eg;
    s_ke[d] = (kraw * rsk) * eg;
    s_v[d]  = vval;
    __syncthreads();

    // ---------------- Phase 2: W(16x128) x S(128x128) via WMMA f32 ----------
    // wave `wave` owns k-rows [wave*32, wave*32+32): 8 K-chunks of 4.
    // A fragment (16x4 f32): lane hi*16+m holds A[m, 2*hi], A[m, 2*hi+1]
    //   row m=0 -> qe, row m=1 -> ke, rows 2..15 -> 0
    // B fragment (4x16 f32): lane hi*16+n holds S[K0+2*hi, col], S[K0+2*hi+1, col]
    // D (16x16 f32, 8 VGPRs): lane n (n<16), VGPR j = D[M=j, N=n]
    const float* S  = ssm_state + (size_t)bh * (DD * DD);
    const int m  = lane & 15;
    const int hi = lane >> 4;

    v8f acc[8] = {};   // 8 N-tiles of 16 columns

    #pragma unroll
    for (int c = 0; c < 8; ++c) {
        const int K0 = wave * 32 + c * 4;

        float a0 = 0.0f, a1 = 0.0f;
        if (m == 0)      { a0 = s_qe[K0 + 2*hi]; a1 = s_qe[K0 + 2*hi + 1]; }
        else if (m == 1) { a0 = s_ke[K0 + 2*hi]; a1 = s_ke[K0 + 2*hi + 1]; }
        v2f A; A.x = a0; A.y = a1;

        const float* rowp = S + (size_t)(K0 + 2*hi) * DD + m;
        #pragma unroll
        for (int t = 0; t < 8; ++t) {
            v2f Bf;
            Bf.x = rowp[t*16];        // S[K0+2*hi,   t*16+m]
            Bf.y = rowp[t*16 + DD];   // S[K0+2*hi+1, t*16+m]
            acc[t] = __builtin_amdgcn_wmma_f32_16x16x4_f32(
                         /*neg_a=*/false, A, /*neg_b=*/false, Bf,
                         /*c_mod=*/(short)0, acc[t],
                         /*reuse_a=*/false, /*reuse_b=*/false);
        }
    }

    // rows M=0 (qS) and M=1 (kS) live in acc[t][0], acc[t][1] of lanes 0..15
    if (hi == 0) {
        #pragma unroll
        for (int t = 0; t < 8; ++t) {
            s_qS[wave][t*16 + m] = acc[t][0];
            s_kS[wave][t*16 + m] = acc[t][1];
        }
    }
    __syncthreads();

    // ---------------- Epilogue: combine partials, rank-1 correction ---------
    const float qS = s_qS[0][d] + s_qS[1][d] + s_qS[2][d] + s_qS[3][d];
    const float kS = s_kS[0][d] + s_kS[1][d] + s_kS[2][d] + s_kS[3][d];

    const float beta_s = fast_sigmoid(beta[(size_t)b * HV + h]);
    const float delta  = (s_v[d] - kS) * beta_s;
    out[(size_t)bh * DD + d] = qS + qk_dot * delta;
}

extern "C" void kernel_launch(void* const* d_in, const int* in_sizes, int n_in,
                              void* d_out, int out_size, void* d_ws, size_t ws_size,
                              hipStream_t stream)
{
    const float* mixed_qkv    = (const float*)d_in[0];
    const float* forget_gate  = (const float*)d_in[1];
    const float* beta         = (const float*)d_in[2];
    const float* conv_state   = (const float*)d_in[3];
    const float* conv_weights = (const float*)d_in[4];
    const float* ssm_state    = (const float*)d_in[5];
    const float* A_log        = (const float*)d_in[6];
    const float* dt_bias      = (const float*)d_in[7];
    float* out = (float*)d_out;

    dim3 grid(BB * HV);   // 4096 blocks, one per (b,h)
    dim3 block(128);      // 4 wave32
    kda_decode_kernel<<<grid, block, 0, stream>>>(
        mixed_qkv, forget_gate, beta, conv_state, conv_weights,
        ssm_state, A_log, dt_bias, out);
}